// LinearAttention_70274254897273
// MI455X (gfx1250) — compile-verified
//
#include <hip/hip_runtime.h>
#include <cstdint>

typedef __attribute__((ext_vector_type(2))) float v2f;
typedef __attribute__((ext_vector_type(8))) float v8f;

#define B_      8
#define L_      4096
#define S_      4096
#define H_      8
#define D_      64
#define DV_     64
#define EPS_    1e-6f
#define SPLIT_  32
#define CHUNK_  (S_ / SPLIT_)   /* 128 rows of S per workgroup */
#define TS_     32              /* rows staged in LDS per stage */
#define NSTAGE_ (CHUNK_ / TS_)  /* 4 */

__device__ __forceinline__ float elup1(float x) {
    // elu(x) + 1 : x>0 -> x+1 ; x<=0 -> exp(x)
    return x > 0.0f ? x + 1.0f : __expf(x);
}

// ---------------------------------------------------------------------------
// Workspace zeroing (KV + Ksum accumulators must start at 0 every call)
// ---------------------------------------------------------------------------
__global__ void la_zero(float* __restrict__ p, int n) {
    int i = blockIdx.x * blockDim.x + threadIdx.x;
    if (i < n) p[i] = 0.0f;
}

// ---------------------------------------------------------------------------
// Phase A: KV[bh][d][dv] = sum_s phi(K)[s,d] * (V[s,dv]/S)
//          Ksum[bh][d]   = sum_s phi(K)[s,d]
// grid: (SPLIT_, B*H), block: 256 (8 waves)
// ---------------------------------------------------------------------------
__global__ __launch_bounds__(256) void la_kv_ksum(const float* __restrict__ keys,
                                                  const float* __restrict__ values,
                                                  float* __restrict__ kv,
                                                  float* __restrict__ ksum) {
    __shared__ float Klds[TS_][D_];
    __shared__ float Vlds[TS_][DV_];
    __shared__ float ksred[256];

    const int t   = threadIdx.x;
    const int bh  = blockIdx.y;
    const int b   = bh / H_;
    const int h   = bh % H_;
    const int s0c = blockIdx.x * CHUNK_;

    const int wave = t >> 5;
    const int lid  = t & 31;
    const int mm   = lid & 15;   // M (or N) index within tile
    const int half = lid >> 4;   // selects K pair {0,1} vs {2,3}

    const int m0 = (wave >> 1) * 16;   // d-tile row (4 rows x 2 waves each)
    const int n0 = (wave & 1) * 32;    // dv-tile col pair

    v8f c0 = {};
    v8f c1 = {};
    float ksacc = 0.0f;
    const int dcol = t & 63;
    const int qrow = (t >> 6) * 8;
    const float inv_s = 1.0f / (float)S_;

    for (int stage = 0; stage < NSTAGE_; ++stage) {
        const int sbase = s0c + stage * TS_;
        // stage 32 rows x 64 cols of K (elu+1) and V (scaled), fully coalesced
#pragma unroll
        for (int i = 0; i < 8; ++i) {
            int idx = t + i * 256;
            int r   = idx >> 6;
            int cc  = idx & 63;
            size_t g = ((size_t)(b * S_ + sbase + r) * H_ + h) * (size_t)D_ + cc;
            Klds[r][cc] = elup1(keys[g]);
            Vlds[r][cc] = values[g] * inv_s;
        }
        __syncthreads();

        // Ksum partials: thread t sums 8 staged rows of its column
#pragma unroll
        for (int r = 0; r < 8; ++r) ksacc += Klds[qrow + r][dcol];

        // 8 WMMA K-steps over the 32 staged rows
#pragma unroll
        for (int ks = 0; ks < TS_; ks += 4) {
            const int kb = ks + half * 2;
            v2f a, bA, bB;
            a.x  = Klds[kb][m0 + mm];          // A[M=d, K=s]
            a.y  = Klds[kb + 1][m0 + mm];
            bA.x = Vlds[kb][n0 + mm];          // B[K=s, N=dv]
            bA.y = Vlds[kb + 1][n0 + mm];
            bB.x = Vlds[kb][n0 + 16 + mm];
            bB.y = Vlds[kb + 1][n0 + 16 + mm];
            c0 = __builtin_amdgcn_wmma_f32_16x16x4_f32(false, a, false, bA, (short)0, c0, false, false);
            c1 = __builtin_amdgcn_wmma_f32_16x16x4_f32(false, a, false, bB, (short)0, c1, false, false);
        }
        __syncthreads();
    }

    // reduce the 4 per-column Ksum partials and merge globally
    ksred[t] = ksacc;
    __syncthreads();
    if (t < 64) {
        float tot = ksred[t] + ksred[t + 64] + ksred[t + 128] + ksred[t + 192];
        atomicAdd(&ksum[(size_t)bh * 64 + t], tot);
    }

    // merge KV partial tiles (C layout: vgpr j -> M = j + 8*half, N = mm)
    float* kvbh = kv + (size_t)bh * 64 * 64;
#pragma unroll
    for (int j = 0; j < 8; ++j) {
        int d = m0 + j + half * 8;
        atomicAdd(&kvbh[d * 64 + n0 + mm],      c0[j]);
        atomicAdd(&kvbh[d * 64 + n0 + 16 + mm], c1[j]);
    }
}

// ---------------------------------------------------------------------------
// Phase B: out[b,l,h,:] = (phi(Q)[l,:] @ KV) * S / (phi(Q)[l,:].Ksum + eps)
// grid: (L/32, B*H), block: 256 (8 waves, one 16x16 tile each -> 32x64)
// ---------------------------------------------------------------------------
__global__ __launch_bounds__(256) void la_out(const float* __restrict__ queries,
                                              const float* __restrict__ kv,
                                              const float* __restrict__ ksum,
                                              float* __restrict__ out) {
    __shared__ float Qlds[32][66];   // stride 66 -> conflict-free strided A reads
    __shared__ float KVlds[64][64];
    __shared__ float Kslds[64];
    __shared__ float zsc[32];
    __shared__ float zpart[32][8];

    const int t  = threadIdx.x;
    const int bh = blockIdx.y;
    const int b  = bh / H_;
    const int h  = bh % H_;
    const int l0 = blockIdx.x * 32;

    // KV block + Ksum into LDS (reused by 128 blocks per head -> L2 resident)
    const float* kvbh = kv + (size_t)bh * 4096;
#pragma unroll
    for (int i = 0; i < 16; ++i) {
        int idx = t + i * 256;
        KVlds[idx >> 6][idx & 63] = kvbh[idx];
    }
    if (t < 64) Kslds[t] = ksum[(size_t)bh * 64 + t];

    // stage 32 Q rows with elu+1 applied once, coalesced
#pragma unroll
    for (int i = 0; i < 8; ++i) {
        int idx = t + i * 256;
        int r   = idx >> 6;
        int cc  = idx & 63;
        size_t g = ((size_t)(b * L_ + l0 + r) * H_ + h) * (size_t)D_ + cc;
        Qlds[r][cc] = elup1(queries[g]);
    }
    __syncthreads();

    // per-row normalizer: z[r] = S / (phi(Q)[r,:].Ksum + eps)
    {
        int r = t >> 3, seg = t & 7;
        float p = 0.0f;
#pragma unroll
        for (int j = 0; j < 8; ++j) p += Qlds[r][seg * 8 + j] * Kslds[seg * 8 + j];
        zpart[r][seg] = p;
    }
    __syncthreads();
    if (t < 32) {
        float zs = 0.0f;
#pragma unroll
        for (int j = 0; j < 8; ++j) zs += zpart[t][j];
        zsc[t] = (float)S_ / (zs + EPS_);
    }
    __syncthreads();

    const int wave = t >> 5;
    const int lid  = t & 31;
    const int mm   = lid & 15;
    const int half = lid >> 4;
    const int m0   = (wave >> 2) * 16;  // 2 tile rows
    const int n0   = (wave & 3) * 16;   // 4 tile cols

    v8f c = {};
#pragma unroll
    for (int k0 = 0; k0 < 64; k0 += 4) {
        const int kb = k0 + half * 2;
        v2f a, bb;
        a.x  = Qlds[m0 + mm][kb];       // A[M=row, K=d]
        a.y  = Qlds[m0 + mm][kb + 1];
        bb.x = KVlds[kb][n0 + mm];      // B[K=d, N=dv]
        bb.y = KVlds[kb + 1][n0 + mm];
        c = __builtin_amdgcn_wmma_f32_16x16x4_f32(false, a, false, bb, (short)0, c, false, false);
    }

    // scale by per-row Z and store (64B contiguous per half-wave)
#pragma unroll
    for (int j = 0; j < 8; ++j) {
        int row = m0 + j + half * 8;
        size_t g = ((size_t)(b * L_ + l0 + row) * H_ + h) * (size_t)DV_ + n0 + mm;
        out[g] = c[j] * zsc[row];
    }
}

// ---------------------------------------------------------------------------
extern "C" void kernel_launch(void* const* d_in, const int* in_sizes, int n_in,
                              void* d_out, int out_size, void* d_ws, size_t ws_size,
                              hipStream_t stream) {
    (void)in_sizes; (void)n_in; (void)out_size; (void)ws_size;
    const float* q = (const float*)d_in[0];
    const float* k = (const float*)d_in[1];
    const float* v = (const float*)d_in[2];
    float* out = (float*)d_out;

    float* kv = (float*)d_ws;                                  // [B*H][64][64]
    float* ks = kv + (size_t)B_ * H_ * 64 * 64;                // [B*H][64]
    const int nz = B_ * H_ * 64 * 64 + B_ * H_ * 64;

    la_zero<<<(nz + 255) / 256, 256, 0, stream>>>(kv, nz);
    la_kv_ksum<<<dim3(SPLIT_, B_ * H_), 256, 0, stream>>>(k, v, kv, ks);
    la_out<<<dim3(L_ / 32, B_ * H_), 256, 0, stream>>>(q, kv, ks, out);
}